// GATA_7464653160732
// MI455X (gfx1250) — compile-verified
//
#include <hip/hip_runtime.h>
#include <hip/hip_bf16.h>

// GATA message-passing layer for MI455X (gfx1250), f32 WMMA (16x16x4) path.
// N=8192 nodes, E=131072 edges, D=128, R=128, H=8, SD=640.

#define NN 8192
#define EE 131072
#define SDIM 640
#define CUTR 5.0f
#define PI_F 3.14159265358979f

typedef __attribute__((ext_vector_type(2))) float v2f;
typedef __attribute__((ext_vector_type(8))) float v8f;

__device__ __forceinline__ float silu_f(float x) { return x / (1.0f + __expf(-x)); }

__device__ __forceinline__ void atomic_max_f32(float* addr, float val) {
  unsigned int* ua = reinterpret_cast<unsigned int*>(addr);
  unsigned int cur = __float_as_uint(*addr);
  while (__uint_as_float(cur) < val) {
    unsigned int prev = atomicCAS(ua, cur, __float_as_uint(val));
    if (prev == cur) break;
    cur = prev;
  }
}

// ---------------------------------------------------------------------------
// WMMA helpers: one wave accumulates a 16x64 tile (4 x v8f accumulators).
// A: row-major (LDS, stride lda), rows r0..r0+15, K = 128.
// B: row-major (global, stride ldb), cols c0..c0+63.
// f32 16x16x4 layout: A frag lane(m=lane&15, kh=(lane>>4)*2) = A[m][k+kh..+1];
// B frag = B[k+kh..+1][c0+t*16+m]; C/D: vgpr r -> row r0 + (lane>>4)*8 + r.
// ---------------------------------------------------------------------------
__device__ __forceinline__ void wmma_acc_16x64(
    const float* __restrict__ A, int lda, int r0,
    const float* __restrict__ B, int ldb, int c0,
    v8f acc[4], int lane)
{
  const int m  = lane & 15;
  const int kh = (lane >> 4) << 1;  // 0 or 2
  const float* Ar = A + (size_t)(r0 + m) * lda + kh;
  const float* Bc = B + c0 + m + (size_t)kh * ldb;
#pragma unroll 4
  for (int kk = 0; kk < 128; kk += 4) {
    v2f a;
    a.x = Ar[kk];
    a.y = Ar[kk + 1];
    const float* b0 = Bc + (size_t)kk * ldb;
#pragma unroll
    for (int t = 0; t < 4; ++t) {
      v2f b;
      b.x = b0[t * 16];
      b.y = b0[t * 16 + ldb];
      acc[t] = __builtin_amdgcn_wmma_f32_16x16x4_f32(
          false, a, false, b, (short)0, acc[t], false, false);
    }
  }
}

__device__ __forceinline__ void init_tile(v8f acc[4], const float* bias, int c0, int lane) {
  const int m = lane & 15;
#pragma unroll
  for (int t = 0; t < 4; ++t) {
    const float bv = bias ? bias[c0 + t * 16 + m] : 0.0f;
#pragma unroll
    for (int r = 0; r < 8; ++r) acc[t][r] = bv;
  }
}

template <bool ACT>
__device__ __forceinline__ void store_tile(float* __restrict__ Out, int ldo,
                                           int r0, int c0, const v8f acc[4], int lane)
{
  const int m  = lane & 15;
  const int mb = r0 + ((lane >> 4) << 3);
#pragma unroll
  for (int t = 0; t < 4; ++t) {
    const int c = c0 + t * 16 + m;
#pragma unroll
    for (int r = 0; r < 8; ++r) {
      float x = acc[t][r];
      if (ACT) x = silu_f(x);
      Out[(size_t)(mb + r) * ldo + c] = x;
    }
  }
}

// 32-row block GEMM: Out[32 x ncols] = act(A[32x128] @ B[128 x ncols] + bias).
// A is LDS (stride 132). 8 waves cooperate; tiles striped across waves.
template <bool ACT>
__device__ __forceinline__ void gemm32(const float* A, const float* B, const float* bias,
                                       float* Out, int ldo, int ncols, int wave, int lane)
{
  const int ntiles = 2 * (ncols >> 6);
  for (int tt = wave; tt < ntiles; tt += 8) {
    const int r0 = (tt & 1) * 16;
    const int c0 = (tt >> 1) * 64;
    v8f acc[4];
    init_tile(acc, bias, c0, lane);
    wmma_acc_16x64(A, 132, r0, B, ncols, c0, acc, lane);
    store_tile<ACT>(Out, ldo, r0, c0, acc, lane);
  }
}

// ---------------------------------------------------------------------------
// Kernel 0: seed outputs (h_out=h, X1_out=X1, X2_out=X2), softmax buffers.
// ---------------------------------------------------------------------------
__global__ __launch_bounds__(256) void k_init(
    const float* __restrict__ h, const float* __restrict__ X1,
    const float* __restrict__ X2,
    float* __restrict__ hout, float* __restrict__ x1out, float* __restrict__ x2out,
    float* __restrict__ m_node, float* __restrict__ den)
{
  size_t i = (size_t)blockIdx.x * 256 + threadIdx.x;
  const size_t nh = (size_t)NN * 128;
  const size_t n1 = (size_t)NN * 384;
  const size_t n2 = (size_t)NN * 640;
  const size_t nm = (size_t)NN * 8;
  if (i < nh) { hout[i] = h[i]; return; }
  i -= nh;
  if (i < n1) { x1out[i] = X1[i]; return; }
  i -= n1;
  if (i < n2) { x2out[i] = X2[i]; return; }
  i -= n2;
  if (i < nm) { m_node[i] = -3.0e38f; return; }
  i -= nm;
  if (i < nm) { den[i] = 0.0f; }
}

// ---------------------------------------------------------------------------
// Kernel 1: per-node GEMMs. 32 nodes / block, 256 threads (8 waves).
//   s = msW2(silu(h@msW1+b1))+b2 ; hn = LN(h) ; q = hn@wq ; k = hn@wk ;
//   v = mvW2(silu(hn@mvW1+b1))+b2
// ---------------------------------------------------------------------------
__global__ __launch_bounds__(256) void k_node(
    const float* __restrict__ h,
    const float* __restrict__ ln_g, const float* __restrict__ ln_b,
    const float* __restrict__ wq, const float* __restrict__ wk,
    const float* __restrict__ mvW1, const float* __restrict__ mvb1,
    const float* __restrict__ mvW2, const float* __restrict__ mvb2,
    const float* __restrict__ msW1, const float* __restrict__ msb1,
    const float* __restrict__ msW2, const float* __restrict__ msb2,
    float* __restrict__ qo, float* __restrict__ ko,
    float* __restrict__ vo, float* __restrict__ so)
{
  __shared__ float lA[32 * 132];
  __shared__ float lB[32 * 132];
  const int tid = threadIdx.x;
  const int wave = tid >> 5, lane = tid & 31;
  const size_t row0 = (size_t)blockIdx.x * 32;

  for (int idx = tid; idx < 32 * 128; idx += 256)
    lA[(idx >> 7) * 132 + (idx & 127)] = h[row0 * 128 + idx];
  __syncthreads();

  // s-branch uses raw h
  gemm32<true>(lA, msW1, msb1, lB, 132, 128, wave, lane);
  __syncthreads();
  gemm32<false>(lB, msW2, msb2, so + row0 * SDIM, SDIM, SDIM, wave, lane);
  __syncthreads();

  // LayerNorm h -> hn (lB). Each wave handles 4 rows; lane covers 4 columns.
#pragma unroll
  for (int rr = 0; rr < 4; ++rr) {
    const int row = wave * 4 + rr;
    const int c = lane * 4;
    float x[4];
#pragma unroll
    for (int j = 0; j < 4; ++j) x[j] = lA[row * 132 + c + j];
    float sum = x[0] + x[1] + x[2] + x[3];
#pragma unroll
    for (int off = 16; off > 0; off >>= 1) sum += __shfl_xor(sum, off, 32);
    const float mean = sum * (1.0f / 128.0f);
    float vs = 0.0f;
#pragma unroll
    for (int j = 0; j < 4; ++j) { x[j] -= mean; vs += x[j] * x[j]; }
#pragma unroll
    for (int off = 16; off > 0; off >>= 1) vs += __shfl_xor(vs, off, 32);
    const float inv = rsqrtf(vs * (1.0f / 128.0f) + 1e-5f);
#pragma unroll
    for (int j = 0; j < 4; ++j)
      lB[row * 132 + c + j] = x[j] * inv * ln_g[c + j] + ln_b[c + j];
  }
  __syncthreads();

  // q and k (8 tiles exactly = 1 per wave), no bias
  {
    const int isK = wave >> 2;
    const int t4 = wave & 3;
    const int r0 = (t4 & 1) * 16;
    const int c0 = (t4 >> 1) * 64;
    const float* B = isK ? wk : wq;
    float* O = (isK ? ko : qo) + row0 * 128;
    v8f acc[4];
    init_tile(acc, nullptr, c0, lane);
    wmma_acc_16x64(lB, 132, r0, B, 128, c0, acc, lane);
    store_tile<false>(O, 128, r0, c0, acc, lane);
  }

  // v-branch from hn
  gemm32<true>(lB, mvW1, mvb1, lA, 132, 128, wave, lane);
  __syncthreads();
  gemm32<false>(lA, mvW2, mvb2, vo + row0 * SDIM, SDIM, SDIM, wave, lane);
}

// ---------------------------------------------------------------------------
// Kernel 2: eq = Xcat@wvq, ek = [X1@wvk1 ; X2@wvk2], node-major (N,8,128).
// 16 nodes / block; X1 rows (16*3=48) then X2 rows (16*5=80) staged in LDS.
// ---------------------------------------------------------------------------
__global__ __launch_bounds__(256) void k_eqek(
    const float* __restrict__ X1, const float* __restrict__ X2,
    const float* __restrict__ wvq, const float* __restrict__ wvk1,
    const float* __restrict__ wvk2,
    float* __restrict__ eq, float* __restrict__ ek)
{
  __shared__ float lA[80 * 132];
  const int tid = threadIdx.x, wave = tid >> 5, lane = tid & 31;
  const size_t nb = (size_t)blockIdx.x * 16;

  // ---- X1 part (rows s=0..2) ----
  for (int idx = tid; idx < 48 * 128; idx += 256)
    lA[(idx >> 7) * 132 + (idx & 127)] = X1[nb * 384 + idx];
  __syncthreads();
  for (int tt = wave; tt < 12; tt += 8) {
    const int isk = (tt >= 6) ? 1 : 0;
    const int t6 = tt - 6 * isk;
    const int r0 = (t6 % 3) * 16;
    const int c0 = (t6 / 3) * 64;
    const float* B = isk ? wvk1 : wvq;
    float* O = isk ? ek : eq;
    v8f acc[4];
    init_tile(acc, nullptr, c0, lane);
    wmma_acc_16x64(lA, 132, r0, B, 128, c0, acc, lane);
    const int m = lane & 15, mb = r0 + ((lane >> 4) << 3);
#pragma unroll
    for (int t = 0; t < 4; ++t) {
      const int c = c0 + t * 16 + m;
#pragma unroll
      for (int r = 0; r < 8; ++r) {
        const int lr = mb + r;
        const size_t node = nb + (size_t)(lr / 3);
        O[node * 1024 + (size_t)(lr % 3) * 128 + c] = acc[t][r];
      }
    }
  }
  __syncthreads();

  // ---- X2 part (rows s=3..7) ----
  for (int idx = tid; idx < 80 * 128; idx += 256)
    lA[(idx >> 7) * 132 + (idx & 127)] = X2[nb * 640 + idx];
  __syncthreads();
  for (int tt = wave; tt < 20; tt += 8) {
    const int isk = (tt >= 10) ? 1 : 0;
    const int t10 = tt - 10 * isk;
    const int r0 = (t10 % 5) * 16;
    const int c0 = (t10 / 5) * 64;
    const float* B = isk ? wvk2 : wvq;
    float* O = isk ? ek : eq;
    v8f acc[4];
    init_tile(acc, nullptr, c0, lane);
    wmma_acc_16x64(lA, 132, r0, B, 128, c0, acc, lane);
    const int m = lane & 15, mb = r0 + ((lane >> 4) << 3);
#pragma unroll
    for (int t = 0; t < 4; ++t) {
      const int c = c0 + t * 16 + m;
#pragma unroll
      for (int r = 0; r < 8; ++r) {
        const int lr = mb + r;
        const size_t node = nb + (size_t)(lr / 5);
        O[node * 1024 + (size_t)(3 + (lr % 5)) * 128 + c] = acc[t][r];
      }
    }
  }
}

// ---------------------------------------------------------------------------
// Kernel 3: per-edge phase 1. 32 edges / block.
//   w = sum_s eq[i]*ek[j]; dtw = mw-MLP(w); dtt = mt-MLP(t); t2 = t + dtw*dtt;
//   re = silu(t2@wre); a[e,h] = sum_d q_i*k_j*re; atomic max into m_node.
// ---------------------------------------------------------------------------
__global__ __launch_bounds__(256) void k_edge1(
    const float* __restrict__ t_ij, const int* __restrict__ edge_index,
    const float* __restrict__ eq, const float* __restrict__ ek,
    const float* __restrict__ mwW1, const float* __restrict__ mwb1,
    const float* __restrict__ mwW2, const float* __restrict__ mwb2,
    const float* __restrict__ mtW1, const float* __restrict__ mtb1,
    const float* __restrict__ mtW2, const float* __restrict__ mtb2,
    const float* __restrict__ wre,
    const float* __restrict__ q, const float* __restrict__ k,
    float* __restrict__ t2o, float* __restrict__ a_buf,
    float* __restrict__ m_node)
{
  __shared__ float lW[32 * 132];
  __shared__ float lT[32 * 132];
  __shared__ float lM[32 * 132];
  __shared__ int sni[32], snj[32];
  const int tid = threadIdx.x, wave = tid >> 5, lane = tid & 31;
  const size_t e0 = (size_t)blockIdx.x * 32;

  if (tid < 32) {
    snj[tid] = edge_index[e0 + tid];        // n_j = edge_index[0]
    sni[tid] = edge_index[EE + e0 + tid];   // n_i = edge_index[1]
  }
  for (int idx = tid; idx < 32 * 128; idx += 256)
    lT[(idx >> 7) * 132 + (idx & 127)] = t_ij[e0 * 128 + idx];
  __syncthreads();

  // w_ij (gather-dot over 8 channel rows; eq/ek are L2-resident tables)
  for (int idx = tid; idx < 32 * 32; idx += 256) {
    const int e = idx >> 5, c4 = idx & 31;
    const float4* pq = reinterpret_cast<const float4*>(eq + (size_t)sni[e] * 1024) + c4;
    const float4* pk = reinterpret_cast<const float4*>(ek + (size_t)snj[e] * 1024) + c4;
    float4 accv = make_float4(0.f, 0.f, 0.f, 0.f);
#pragma unroll
    for (int s = 0; s < 8; ++s) {
      const float4 aq = pq[s * 32], bk = pk[s * 32];
      accv.x += aq.x * bk.x; accv.y += aq.y * bk.y;
      accv.z += aq.z * bk.z; accv.w += aq.w * bk.w;
    }
    float* dst = lW + e * 132 + c4 * 4;
    dst[0] = accv.x; dst[1] = accv.y; dst[2] = accv.z; dst[3] = accv.w;
  }
  __syncthreads();

  gemm32<true>(lW, mwW1, mwb1, lM, 132, 128, wave, lane);   // mid_w
  __syncthreads();
  gemm32<false>(lM, mwW2, mwb2, lW, 132, 128, wave, lane);  // dt_w -> lW
  __syncthreads();
  gemm32<true>(lT, mtW1, mtb1, lM, 132, 128, wave, lane);   // mid_t
  __syncthreads();

  // dt_t GEMM with fused t2 = t + dt_w * dt_t epilogue
  for (int tt = wave; tt < 4; tt += 8) {
    const int r0 = (tt & 1) * 16, c0 = (tt >> 1) * 64;
    v8f acc[4];
    init_tile(acc, mtb2, c0, lane);
    wmma_acc_16x64(lM, 132, r0, mtW2, 128, c0, acc, lane);
    const int m = lane & 15, mb = r0 + ((lane >> 4) << 3);
#pragma unroll
    for (int t = 0; t < 4; ++t) {
      const int c = c0 + t * 16 + m;
#pragma unroll
      for (int r = 0; r < 8; ++r) {
        const int row = mb + r;
        const float t2v = lT[row * 132 + c] + lW[row * 132 + c] * acc[t][r];
        lT[row * 132 + c] = t2v;
        t2o[(e0 + row) * 128 + c] = t2v;
      }
    }
  }
  __syncthreads();

  gemm32<true>(lT, wre, nullptr, lM, 132, 128, wave, lane); // silu(t2@wre)
  __syncthreads();

  // attention logits, 1 thread = 1 (edge, head)
  {
    const int e = tid >> 3, hh = tid & 7;
    const float* qp = q + (size_t)sni[e] * 128 + hh * 16;
    const float* kp = k + (size_t)snj[e] * 128 + hh * 16;
    const float* rp = lM + e * 132 + hh * 16;
    float a = 0.0f;
#pragma unroll
    for (int d = 0; d < 16; ++d) a += qp[d] * kp[d] * rp[d];
    a_buf[(e0 + e) * 8 + hh] = a;
    atomic_max_f32(m_node + (size_t)sni[e] * 8 + hh, a);
  }
}

// ---------------------------------------------------------------------------
// Kernel 4: softmax numerator + denominator accumulation.
// ---------------------------------------------------------------------------
__global__ __launch_bounds__(256) void k_edge2(
    const int* __restrict__ edge_index,
    const float* __restrict__ m_node, float* __restrict__ a_buf,
    float* __restrict__ den)
{
  const size_t idx = (size_t)blockIdx.x * 256 + threadIdx.x;
  const size_t e = idx >> 3;
  const int hh = (int)(idx & 7);
  const int ni = edge_index[EE + e];
  const float ex = __expf(a_buf[idx] - m_node[(size_t)ni * 8 + hh]);
  a_buf[idx] = ex;
  atomicAdd(den + (size_t)ni * 8 + hh, ex);
}

// ---------------------------------------------------------------------------
// Kernel 5: final. rs = t2@wrs (fused); o = alpha*v_j + rs*s_j*cut;
// scatter into h_out / X1_out / X2_out with atomics.
// ---------------------------------------------------------------------------
__global__ __launch_bounds__(256) void k_edge3(
    const int* __restrict__ edge_index,
    const float* __restrict__ t2o, const float* __restrict__ wrs,
    const float* __restrict__ vtab, const float* __restrict__ stab,
    const float* __restrict__ a_buf, const float* __restrict__ den,
    const float* __restrict__ r0in, const float* __restrict__ r1in,
    const float* __restrict__ r2in,
    const float* __restrict__ X1, const float* __restrict__ X2,
    float* __restrict__ hout, float* __restrict__ x1out, float* __restrict__ x2out)
{
  __shared__ float lT[32 * 132];
  __shared__ float lAl[32 * 8];
  __shared__ float lcut[32];
  __shared__ int sni[32], snj[32];
  const int tid = threadIdx.x, wave = tid >> 5, lane = tid & 31;
  const size_t e0 = (size_t)blockIdx.x * 32;

  if (tid < 128) __builtin_prefetch(wrs + (size_t)tid * SDIM, 0, 3);
  if (tid < 32) {
    snj[tid] = edge_index[e0 + tid];
    sni[tid] = edge_index[EE + e0 + tid];
    const float r0v = r0in[e0 + tid];
    lcut[tid] = (r0v <= CUTR) ? 0.5f * (cosf(PI_F * r0v / CUTR) + 1.0f) : 0.0f;
  }
  for (int idx = tid; idx < 32 * 128; idx += 256)
    lT[(idx >> 7) * 132 + (idx & 127)] = t2o[e0 * 128 + idx];
  __syncthreads();
  {
    const int e = tid >> 3, hh = tid & 7;
    lAl[tid] = a_buf[(e0 + e) * 8 + hh] / den[(size_t)sni[e] * 8 + hh];
  }
  __syncthreads();

  for (int tt = wave; tt < 20; tt += 8) {
    const int r0 = (tt & 1) * 16;
    const int c0 = (tt >> 1) * 64;
    v8f acc[4];
    init_tile(acc, nullptr, c0, lane);
    wmma_acc_16x64(lT, 132, r0, wrs, SDIM, c0, acc, lane);
    const int m = lane & 15, mb = r0 + ((lane >> 4) << 3);
#pragma unroll
    for (int t = 0; t < 4; ++t) {
      const int f = c0 + t * 16 + m;     // 0..639, uniform per lane
      const int seg = f >> 7;            // which 128-slice of o
      const int d = f & 127;
      const int hh = f / 80;             // head for alpha
#pragma unroll
      for (int r = 0; r < 8; ++r) {
        const int e = mb + r;
        const size_t eg = e0 + e;
        const size_t ni_ = (size_t)sni[e];
        const size_t nj_ = (size_t)snj[e];
        const float o = lAl[e * 8 + hh] * vtab[nj_ * SDIM + f]
                      + acc[t][r] * stab[nj_ * SDIM + f] * lcut[e];
        if (seg == 0) {
          atomicAdd(hout + ni_ * 128 + d, o);
        } else if (seg == 1) {
#pragma unroll
          for (int c = 0; c < 3; ++c)
            atomicAdd(x1out + ni_ * 384 + (size_t)c * 128 + d, o * r1in[eg * 3 + c]);
        } else if (seg == 2) {
#pragma unroll
          for (int c = 0; c < 5; ++c)
            atomicAdd(x2out + ni_ * 640 + (size_t)c * 128 + d, o * r2in[eg * 5 + c]);
        } else if (seg == 3) {
#pragma unroll
          for (int c = 0; c < 3; ++c)
            atomicAdd(x1out + ni_ * 384 + (size_t)c * 128 + d,
                      o * X1[nj_ * 384 + (size_t)c * 128 + d]);
        } else {
#pragma unroll
          for (int c = 0; c < 5; ++c)
            atomicAdd(x2out + ni_ * 640 + (size_t)c * 128 + d,
                      o * X2[nj_ * 640 + (size_t)c * 128 + d]);
        }
      }
    }
  }
}

// ---------------------------------------------------------------------------
extern "C" void kernel_launch(void* const* d_in, const int* in_sizes, int n_in,
                              void* d_out, int out_size, void* d_ws, size_t ws_size,
                              hipStream_t stream)
{
  const float* h    = (const float*)d_in[0];
  const float* X1   = (const float*)d_in[1];
  const float* X2   = (const float*)d_in[2];
  const float* t_ij = (const float*)d_in[3];
  const float* r0   = (const float*)d_in[4];
  const float* r1   = (const float*)d_in[5];
  const float* r2   = (const float*)d_in[6];
  const int*   eidx = (const int*)d_in[7];
  const float* wvq  = (const float*)d_in[8];
  const float* wvk1 = (const float*)d_in[9];
  const float* wvk2 = (const float*)d_in[10];
  const float* mwW1 = (const float*)d_in[11];
  const float* mwb1 = (const float*)d_in[12];
  const float* mwW2 = (const float*)d_in[13];
  const float* mwb2 = (const float*)d_in[14];
  const float* mtW1 = (const float*)d_in[15];
  const float* mtb1 = (const float*)d_in[16];
  const float* mtW2 = (const float*)d_in[17];
  const float* mtb2 = (const float*)d_in[18];
  const float* ln_g = (const float*)d_in[19];
  const float* ln_b = (const float*)d_in[20];
  const float* wq   = (const float*)d_in[21];
  const float* wk   = (const float*)d_in[22];
  const float* mvW1 = (const float*)d_in[23];
  const float* mvb1 = (const float*)d_in[24];
  const float* mvW2 = (const float*)d_in[25];
  const float* mvb2 = (const float*)d_in[26];
  const float* wre  = (const float*)d_in[27];
  const float* wrs  = (const float*)d_in[28];
  const float* msW1 = (const float*)d_in[29];
  const float* msb1 = (const float*)d_in[30];
  const float* msW2 = (const float*)d_in[31];
  const float* msb2 = (const float*)d_in[32];

  float* out   = (float*)d_out;
  float* hout  = out;
  float* x1out = hout + (size_t)NN * 128;
  float* x2out = x1out + (size_t)NN * 384;
  float* t2out = x2out + (size_t)NN * 640;

  float* ws   = (float*)d_ws;
  float* qbuf = ws; ws += (size_t)NN * 128;
  float* kbuf = ws; ws += (size_t)NN * 128;
  float* vbuf = ws; ws += (size_t)NN * 640;
  float* sbuf = ws; ws += (size_t)NN * 640;
  float* eqb  = ws; ws += (size_t)NN * 1024;
  float* ekb  = ws; ws += (size_t)NN * 1024;
  float* abuf = ws; ws += (size_t)EE * 8;
  float* mbuf = ws; ws += (size_t)NN * 8;
  float* dbuf = ws; ws += (size_t)NN * 8;

  const size_t tot_init = (size_t)NN * (128 + 384 + 640 + 8 + 8);
  k_init<<<(int)((tot_init + 255) / 256), 256, 0, stream>>>(
      h, X1, X2, hout, x1out, x2out, mbuf, dbuf);
  k_node<<<NN / 32, 256, 0, stream>>>(
      h, ln_g, ln_b, wq, wk, mvW1, mvb1, mvW2, mvb2,
      msW1, msb1, msW2, msb2, qbuf, kbuf, vbuf, sbuf);
  k_eqek<<<NN / 16, 256, 0, stream>>>(X1, X2, wvq, wvk1, wvk2, eqb, ekb);
  k_edge1<<<EE / 32, 256, 0, stream>>>(
      t_ij, eidx, eqb, ekb, mwW1, mwb1, mwW2, mwb2,
      mtW1, mtb1, mtW2, mtb2, wre, qbuf, kbuf, t2out, abuf, mbuf);
  k_edge2<<<(EE * 8) / 256, 256, 0, stream>>>(eidx, mbuf, abuf, dbuf);
  k_edge3<<<EE / 32, 256, 0, stream>>>(
      eidx, t2out, wrs, vbuf, sbuf, abuf, dbuf, r0, r1, r2, X1, X2,
      hout, x1out, x2out);

  (void)in_sizes; (void)n_in; (void)out_size; (void)ws_size;
}